// CompositionalEmbeddings_5274219839685
// MI455X (gfx1250) — compile-verified
//
#include <hip/hip_runtime.h>
#include <stdint.h>

// CompositionalEmbeddings for MI455X (gfx1250).
//
// 0-FLOP gather: out[t, 0:512]    = token_table[id[t]]
//                out[t, 512:1024] = concat(op,var,const,struct,special)[id[t]]
// ~512 MB HBM traffic => ~22 us floor @ 23.3 TB/s. Data path: CDNA5 async copy
// engine (global -> LDS -> global, ASYNCcnt-tracked), GVS addressing so row
// bases live in SGPR pairs and chunk offsets are instruction immediates
// (applied to both the global and LDS side). Output stores are non-temporal so
// the 256 MB write stream doesn't evict the ~200 MB table set from the 192 MB L2.

#define HALF 512
#define WAVES_PER_BLOCK 8

// copies 16B/lane from global address sbase + voff + OFF to the LDS address in ldsv plus OFF, ASYNCcnt-tracked.
#define ASYNC_LD(ldsv, voff, sbase, OFF)                                      \
  asm volatile("global_load_async_to_lds_b128 %0, %1, %2 offset:" #OFF        \
               :: "v"(ldsv), "v"(voff), "s"(sbase) : "memory")

// copies 16B/lane from the LDS address in ldsv plus OFF to global address sbase + voff + OFF, non-temporal store.
#define ASYNC_ST(voff, ldsv, sbase, OFF)                                      \
  asm volatile("global_store_async_from_lds_b128 %0, %1, %2 offset:" #OFF     \
               " th:TH_STORE_NT"                                              \
               :: "v"(voff), "v"(ldsv), "s"(sbase) : "memory")

__device__ __forceinline__ void wait_async_zero() {
  asm volatile("s_wait_asynccnt 0" ::: "memory");
}

__global__ __launch_bounds__(32 * WAVES_PER_BLOCK)
void CompositionalEmbeddings_gather_kernel(const int* __restrict__ token_ids,
                                           const float* __restrict__ token_table,
                                           const float* __restrict__ op_table,
                                           const float* __restrict__ var_table,
                                           const float* __restrict__ const_table,
                                           const float* __restrict__ struct_table,
                                           const float* __restrict__ special_table,
                                           float* __restrict__ out) {
  // 4 KB staging per wave: [0,2048) token-row bytes, [2048,4096) category-row.
  __shared__ float lbuf[WAVES_PER_BLOCK * 1024];

  const int lane = threadIdx.x & 31;
  const int wid  = threadIdx.x >> 5;
  const int t    = blockIdx.x * WAVES_PER_BLOCK + wid;  // token (wave-uniform)

  // id is wave-uniform: force it scalar so row-pointer math runs on the SALU
  // with scalar branches (no EXEC divergence, no VALU magic-division).
  const int id = __builtin_amdgcn_readfirstlane(token_ids[t]);

  const float* trow = token_table + (size_t)id * HALF;

  // Category tables stack to rows 0..49999 of the virtual concat table.
  const float* crow;
  if (id < 40) {
    const float* base;
    int r = id;
    if      (id < 10) { base = op_table; }
    else if (id < 20) { base = var_table;    r = id - 10; }
    else if (id < 30) { base = const_table;  r = id - 20; }
    else              { base = struct_table; r = id - 30; }
    crow = base + (size_t)r * HALF;
  } else {
    crow = special_table + (size_t)(id - 40) * HALF;
  }

  const float* orow = out + (size_t)t * 1024;

  // Flat-address LDS aperture keeps the LDS byte offset in addr[31:0].
  const unsigned lds0  = (unsigned)(uintptr_t)&lbuf[wid * 1024];
  const unsigned voff  = (unsigned)lane * 16u;   // per-lane 16B chunk offset
  const unsigned lds_t = lds0 + voff;            // token half, chunk 0
  const unsigned lds_c = lds_t + 2048u;          // category half, chunk 0

  // Stage both 2KB rows into LDS: 8 async b128 loads (32 lanes x 16B each).
  ASYNC_LD(lds_t, voff, trow, 0);
  ASYNC_LD(lds_t, voff, trow, 512);
  ASYNC_LD(lds_t, voff, trow, 1024);
  ASYNC_LD(lds_t, voff, trow, 1536);
  ASYNC_LD(lds_c, voff, crow, 0);
  ASYNC_LD(lds_c, voff, crow, 512);
  ASYNC_LD(lds_c, voff, crow, 1024);
  ASYNC_LD(lds_c, voff, crow, 1536);

  wait_async_zero();   // all 8 loads have landed in this wave's LDS slice

  // Drain LDS to the contiguous 4KB output row: 8 async b128 NT stores.
  // IOFFSET applies to both sides, so one LDS-address VGPR serves all chunks.
  ASYNC_ST(voff, lds_t, orow, 0);
  ASYNC_ST(voff, lds_t, orow, 512);
  ASYNC_ST(voff, lds_t, orow, 1024);
  ASYNC_ST(voff, lds_t, orow, 1536);
  ASYNC_ST(voff, lds_t, orow, 2048);
  ASYNC_ST(voff, lds_t, orow, 2560);
  ASYNC_ST(voff, lds_t, orow, 3072);
  ASYNC_ST(voff, lds_t, orow, 3584);
  // s_endpgm performs an implicit wait-idle, so the async stores complete
  // before the wave retires.
}

extern "C" void kernel_launch(void* const* d_in, const int* in_sizes, int n_in,
                              void* d_out, int out_size, void* d_ws, size_t ws_size,
                              hipStream_t stream) {
  const int*   token_ids     = (const int*)  d_in[0];
  const float* token_table   = (const float*)d_in[1];
  const float* op_table      = (const float*)d_in[2];
  const float* var_table     = (const float*)d_in[3];
  const float* const_table   = (const float*)d_in[4];
  const float* struct_table  = (const float*)d_in[5];
  const float* special_table = (const float*)d_in[6];
  float* out = (float*)d_out;

  const int n_tokens = in_sizes[0];                 // 32 * 2048 = 65536
  const int blocks   = n_tokens / WAVES_PER_BLOCK;  // one token per wave

  hipLaunchKernelGGL(CompositionalEmbeddings_gather_kernel,
                     dim3(blocks), dim3(32 * WAVES_PER_BLOCK), 0, stream,
                     token_ids, token_table, op_table, var_table, const_table,
                     struct_table, special_table, out);
}